// GCNLayer_33956011442288
// MI455X (gfx1250) — compile-verified
//
#include <hip/hip_runtime.h>
#include <math.h>

// GCN layer: out = D^-1/2 (A+I) D^-1/2 (XW + b), COO edges, N=50000, E=800000, D=128.
//
// MI455X reasoning: the GEMM is only 1.64 GFLOP / ~77 MB -> bandwidth bound (~2us),
// so we keep exact fp32 semantics via V_WMMA_F32_16X16X4_F32 instead of dropping to
// f16. The scatter-add (800k x 512B gathers + 102M f32 atomics, Hs fits in 192MB L2)
// dominates at ~35us; it is implemented exactly like the reference's segment_sum.

#define D 128

typedef __attribute__((ext_vector_type(2))) float v2f;
typedef __attribute__((ext_vector_type(8))) float v8f;

// ---------- degree / normalization ----------
__global__ void k_zero_deg(int* __restrict__ deg, int n) {
    int i = blockIdx.x * blockDim.x + threadIdx.x;
    if (i < n) deg[i] = 0;
}

__global__ void k_count_deg(const int* __restrict__ ecol, int* __restrict__ deg, int e) {
    int i = blockIdx.x * blockDim.x + threadIdx.x;
    if (i < e) atomicAdd(&deg[ecol[i]], 1);
}

__global__ void k_dinv(const int* __restrict__ deg, float* __restrict__ dinv, int n) {
    int i = blockIdx.x * blockDim.x + threadIdx.x;
    if (i < n) dinv[i] = rsqrtf((float)deg[i] + 1.0f);
}

// ---------- GEMM: Hs = (X@W + b) * dinv[row];  out_init = Hs * dinv[row] ----------
// Block = 256 threads = 8 waves; wave w computes rows [blk*128 + 16w, +16).
// W (128x128 f32 = 64KB) + bias staged in LDS (WGP has 320KB).
// WMMA f32 16x16x4 layouts (ISA 7.12.2, 32-bit operands):
//   A 16x4 : lane m=lane%16, half=lane/16; a[v] = A[m][2*half + v]
//   B 4x16 : lane n=lane%16;               b[v] = B[2*half + v][n]
//   C/D    : c[v] = C[v + 8*half][n]
__global__ __launch_bounds__(256)
void k_gemm(const float* __restrict__ X, const float* __restrict__ W,
            const float* __restrict__ bias, const float* __restrict__ dinv,
            float* __restrict__ Hs, float* __restrict__ out, int n) {
    __shared__ float Wl[D * D];
    __shared__ float bl[D];

    const int tid = threadIdx.x;

    // cooperative W load, float4 (4096 float4 / 256 threads = 16 iters)
    {
        const float4* W4 = (const float4*)W;
        float4* Wl4 = (float4*)Wl;
        #pragma unroll
        for (int i = 0; i < (D * D / 4) / 256; ++i)
            Wl4[tid + i * 256] = W4[tid + i * 256];
        if (tid < D) bl[tid] = bias[tid];
    }
    __syncthreads();

    const int wave = tid >> 5;
    const int lane = tid & 31;
    const int half = lane >> 4;   // 0/1
    const int ln16 = lane & 15;
    const int row_base = blockIdx.x * 128 + wave * 16;

    // A fragments for the full K=128 (32 k-steps of 4): per-lane float2 loads.
    v2f a[32];
    int r_load = row_base + ln16;
    if (r_load >= n) r_load = n - 1;   // clamped tail; stores are predicated below
    {
        const float2* x2 = (const float2*)(X + (size_t)r_load * D);
        #pragma unroll
        for (int k = 0; k < 32; ++k) {
            float2 t = x2[2 * k + half];     // element K = 4k + 2*half (+0,+1)
            v2f av; av[0] = t.x; av[1] = t.y;
            a[k] = av;
        }
    }

    // Left-scale factors for the 8 output rows this lane touches (row = base+8*half+v).
    float srow[8];
    #pragma unroll
    for (int v = 0; v < 8; ++v) {
        int r = row_base + 8 * half + v;
        srow[v] = (r < n) ? dinv[r] : 0.0f;
    }

    for (int nt = 0; nt < 8; ++nt) {
        const int col = nt * 16 + ln16;
        v8f acc = {};
        #pragma unroll
        for (int k = 0; k < 32; ++k) {
            const int krow = 4 * k + 2 * half;
            v2f bf;
            bf[0] = Wl[krow * D + col];
            bf[1] = Wl[(krow + 1) * D + col];
            acc = __builtin_amdgcn_wmma_f32_16x16x4_f32(
                /*neg_a=*/false, a[k], /*neg_b=*/false, bf,
                /*c_mod=*/(short)0, acc, /*reuse_a=*/false, /*reuse_b=*/false);
        }
        const float bv = bl[col];
        #pragma unroll
        for (int v = 0; v < 8; ++v) {
            const int r = row_base + 8 * half + v;
            if (r < n) {
                const float s  = srow[v];
                const float hs = (acc[v] + bv) * s;  // right D^-1/2
                const size_t idx = (size_t)r * D + col;
                Hs[idx]  = hs;                        // for the edge gather
                out[idx] = hs * s;                    // self-loop term, left-scaled
            }
        }
    }
}

// ---------- scatter: out[row] += dinv[row] * Hs[col] per edge ----------
// One wave per edge; lane covers 4 channels via float4 gather + 4 f32 atomics.
__global__ __launch_bounds__(256)
void k_scatter(const int* __restrict__ erow, const int* __restrict__ ecol,
               const float* __restrict__ Hs, const float* __restrict__ dinv,
               float* __restrict__ out, int e) {
    const int gtid = blockIdx.x * 256 + threadIdx.x;
    const int edge = gtid >> 5;
    const int lane = gtid & 31;
    if (edge >= e) return;
    const int row = erow[edge];
    const int col = ecol[edge];
    const float s = dinv[row];                 // left D^-1/2 folded into contribution
    const float4 v = ((const float4*)(Hs + (size_t)col * D))[lane];
    float* dst = out + (size_t)row * D + lane * 4;
    atomicAdd(dst + 0, s * v.x);
    atomicAdd(dst + 1, s * v.y);
    atomicAdd(dst + 2, s * v.z);
    atomicAdd(dst + 3, s * v.w);
}

extern "C" void kernel_launch(void* const* d_in, const int* in_sizes, int n_in,
                              void* d_out, int out_size, void* d_ws, size_t ws_size,
                              hipStream_t stream) {
    const float* X    = (const float*)d_in[0];
    const float* W    = (const float*)d_in[1];
    const float* b    = (const float*)d_in[2];
    const int*   erow = (const int*)d_in[3];
    const int*   ecol = (const int*)d_in[4];
    float* out = (float*)d_out;

    const int N = in_sizes[0] / D;
    const int E = in_sizes[3];

    // ws layout: [deg: N int][dinv: N float][Hs: N*D float]  (~26 MB total)
    char* ws = (char*)d_ws;
    int*   deg  = (int*)ws;
    float* dinv = (float*)(ws + (size_t)N * sizeof(int));
    float* Hs   = (float*)(ws + 2 * (size_t)N * sizeof(float));

    k_zero_deg<<<(N + 255) / 256, 256, 0, stream>>>(deg, N);
    k_count_deg<<<(E + 255) / 256, 256, 0, stream>>>(ecol, deg, E);
    k_dinv<<<(N + 255) / 256, 256, 0, stream>>>(deg, dinv, N);

    const int gemm_blocks = (N + 127) / 128;   // 128 rows per 8-wave block
    k_gemm<<<gemm_blocks, 256, 0, stream>>>(X, W, b, dinv, Hs, out, N);

    const long long sthreads = (long long)E * 32;  // one wave32 per edge
    const int scat_blocks = (int)((sthreads + 255) / 256);
    k_scatter<<<scat_blocks, 256, 0, stream>>>(erow, ecol, Hs, dinv, out, E);
}